// Octree2Col_11854109737086
// MI455X (gfx1250) — compile-verified
//
#include <hip/hip_runtime.h>
#include <stdint.h>
#include <stddef.h>

// Octree2Col masked row-gather for MI455X (gfx1250, wave32).
// out[row, 0:C] = neigh[row] >= 0 ? data[neigh[row], 0:C] : 0
//
// Roofline: 0 FLOPs. Output = N*K*C*4B = 1.38 GB streamed once (~59 us at
// 23.3 TB/s HBM). data = 51 MB -> L2-resident (192 MB L2). Gather reads hit
// L2 (RT); output uses non-temporal stores so it can't evict the table.
//
// Fast path (C == 128): one wave32 per row (32 lanes x b128 = 512 B), 4 rows
// per wave through the gfx1250 async global->LDS->global DMA path:
// 4 async RT loads in flight (ASYNCcnt) -> one s_wait_asynccnt -> 4 NT async
// stores. Row indices are readfirstlane'd into SGPRs so validity tests are
// scalar branches and async addressing is saddr-form (shared lane*16 vaddr).

typedef float v4f __attribute__((ext_vector_type(4)));
typedef int   v4i __attribute__((vector_size(16)));   // matches builtin param type

typedef v4i __attribute__((address_space(1)))* gbl_v4i_ptr;  // global (AS1)
typedef v4i __attribute__((address_space(3)))* lds_v4i_ptr;  // LDS    (AS3)

// gfx12-style CPol: TH in bits [2:0]; TH=1 -> non-temporal.
#define OC_CPOL_RT 0
#define OC_CPOL_NT 1

#if __has_builtin(__builtin_amdgcn_global_load_async_to_lds_b128) && \
    __has_builtin(__builtin_amdgcn_global_store_async_from_lds_b128) && \
    __has_builtin(__builtin_amdgcn_s_wait_asynccnt)
#define OC_USE_ASYNC_LDS 1
#else
#define OC_USE_ASYNC_LDS 0
#endif

#define OC_ROWS_PER_WAVE 4
#define OC_WAVES_PER_BLOCK 8

#if OC_USE_ASYNC_LDS
// Specialized for C == 128 (C4 == 32): one full row per b128 async op.
__global__ __launch_bounds__(256) void octree2col_async4_kernel(
    const float* __restrict__ data,
    const int*   __restrict__ neigh,
    float*       __restrict__ out,
    int R)    // total rows = N*K
{
  const int wave = (int)(threadIdx.x >> 5);
  const int lane = (int)(threadIdx.x & 31);
  // Wave-uniform by construction; readfirstlane pins it to an SGPR.
  const int base = __builtin_amdgcn_readfirstlane(
      ((int)blockIdx.x * OC_WAVES_PER_BLOCK + wave) * OC_ROWS_PER_WAVE);
  if (base >= R) return;                      // wave-uniform exit

  __shared__ v4f sbuf[OC_WAVES_PER_BLOCK][OC_ROWS_PER_WAVE][32];  // 16 KB

  const v4f* __restrict__ dat4 = (const v4f*)data;
  v4f* __restrict__ out4 = (v4f*)out;

  if (base + OC_ROWS_PER_WAVE <= R) {
    // ---- Full tile: no per-row bounds checks (the only path when R%32==0).
    int nbs[OC_ROWS_PER_WAVE];
#pragma unroll
    for (int i = 0; i < OC_ROWS_PER_WAVE; ++i)
      nbs[i] = __builtin_amdgcn_readfirstlane(neigh[base + i]);  // SGPR indices

    // Phase 1: up to 4 async b128 row loads in flight (RT: keep data in L2).
#pragma unroll
    for (int i = 0; i < OC_ROWS_PER_WAVE; ++i) {
      if (nbs[i] >= 0) {
        __builtin_amdgcn_global_load_async_to_lds_b128(
            (gbl_v4i_ptr)(dat4 + (size_t)nbs[i] * 32 + lane),
            (lds_v4i_ptr)(&sbuf[wave][i][lane]),
            /*offset=*/0, /*cpol=*/OC_CPOL_RT);
      }
    }
    __builtin_amdgcn_s_wait_asynccnt(0);      // all row loads landed in LDS

    // Phase 2: batched non-temporal stores (write-once 1.38 GB stream).
#pragma unroll
    for (int i = 0; i < OC_ROWS_PER_WAVE; ++i) {
      v4f* dst = out4 + (size_t)(base + i) * 32 + lane;
      if (nbs[i] >= 0) {
        __builtin_amdgcn_global_store_async_from_lds_b128(
            (gbl_v4i_ptr)dst,
            (lds_v4i_ptr)(&sbuf[wave][i][lane]),
            /*offset=*/0, /*cpol=*/OC_CPOL_NT);
      } else {
        const v4f z = {0.0f, 0.0f, 0.0f, 0.0f};
        __builtin_nontemporal_store(z, dst);
      }
    }
  } else {
    // ---- Tail tile (at most one wave in the grid): per-row guards.
    const int nrows = R - base;
    int nbs[OC_ROWS_PER_WAVE];
#pragma unroll
    for (int i = 0; i < OC_ROWS_PER_WAVE; ++i)
      nbs[i] = (i < nrows) ? __builtin_amdgcn_readfirstlane(neigh[base + i]) : -1;

#pragma unroll
    for (int i = 0; i < OC_ROWS_PER_WAVE; ++i) {
      if (i < nrows && nbs[i] >= 0) {
        __builtin_amdgcn_global_load_async_to_lds_b128(
            (gbl_v4i_ptr)(dat4 + (size_t)nbs[i] * 32 + lane),
            (lds_v4i_ptr)(&sbuf[wave][i][lane]),
            /*offset=*/0, /*cpol=*/OC_CPOL_RT);
      }
    }
    __builtin_amdgcn_s_wait_asynccnt(0);

#pragma unroll
    for (int i = 0; i < OC_ROWS_PER_WAVE; ++i) {
      if (i >= nrows) break;
      v4f* dst = out4 + (size_t)(base + i) * 32 + lane;
      if (nbs[i] >= 0) {
        __builtin_amdgcn_global_store_async_from_lds_b128(
            (gbl_v4i_ptr)dst,
            (lds_v4i_ptr)(&sbuf[wave][i][lane]),
            /*offset=*/0, /*cpol=*/OC_CPOL_NT);
      } else {
        const v4f z = {0.0f, 0.0f, 0.0f, 0.0f};
        __builtin_nontemporal_store(z, dst);
      }
    }
  }
  // s_endpgm implicitly waits all counters; stores complete before kernel end.
}
#endif

// Generic fallback (any C divisible by 4): direct b128 gather + NT store,
// one thread per float4 element of the output.
__global__ __launch_bounds__(256) void octree2col_generic_kernel(
    const float* __restrict__ data,
    const int*   __restrict__ neigh,
    float*       __restrict__ out,
    int R, int C4)
{
  const long long total = (long long)R * (long long)C4;
  const long long idx = (long long)blockIdx.x * blockDim.x + threadIdx.x;
  if (idx >= total) return;

  const int row = (int)(idx / C4);
  const int c   = (int)(idx % C4);
  const int nb  = neigh[row];

  v4f v = {0.0f, 0.0f, 0.0f, 0.0f};
  if (nb >= 0) v = ((const v4f*)data)[(size_t)nb * C4 + c];
  __builtin_nontemporal_store(v, (v4f*)out + idx);
}

extern "C" void kernel_launch(void* const* d_in, const int* in_sizes, int n_in,
                              void* d_out, int out_size, void* d_ws, size_t ws_size,
                              hipStream_t stream) {
  (void)n_in; (void)d_ws; (void)ws_size;

  const float* data  = (const float*)d_in[0];   // [N, C] fp32
  const int*   neigh = (const int*)d_in[1];     // [N, K] int
  float*       out   = (float*)d_out;           // [N, K, C] fp32

  const int R  = in_sizes[1];                   // N*K rows
  const int C  = (R > 0) ? (out_size / R) : 0;  // 128 for this problem
  const int C4 = C >> 2;

#if OC_USE_ASYNC_LDS
  if (C4 == 32) {
    const int rows_per_block = OC_WAVES_PER_BLOCK * OC_ROWS_PER_WAVE;  // 32
    const int blocks = (R + rows_per_block - 1) / rows_per_block;
    octree2col_async4_kernel<<<blocks, 256, 0, stream>>>(data, neigh, out, R);
    return;
  }
#endif
  const long long total = (long long)R * (long long)C4;
  const int blocks = (int)((total + 255) / 256);
  octree2col_generic_kernel<<<blocks, 256, 0, stream>>>(data, neigh, out, R, C4);
}